// MeanFieldMultiDimensionalLogisticRegression_4346506904037
// MI455X (gfx1250) — compile-verified
//
#include <hip/hip_runtime.h>
#include <math.h>

// MeanField logistic regression, MI455X (gfx1250), wave32 + WMMA f32 16x16x4.
//
// Inputs (setup_inputs order):
//   d_in[0] = X         [200000, 64] f32
//   d_in[1] = w_mu      [64]         f32
//   d_in[2] = w_log_var [64]         f32
//   d_in[3] = z         [100]        f32
// Output: d_out = [ Y (N*100 f32) | activation_mat (N*100 f32) ]
//
// Roofline: 51.2 MB X reads + 160 MB output stores => HBM-bound (~9 us at
// 23.3 TB/s). Design goal: fully coalesced stores, minimal VALU per output
// element (exp + add + rcp), WMMA for both the K=64 reductions and the
// rank-2 (std,mean)x(z,1) activation tiles.

typedef __attribute__((ext_vector_type(2))) float v2f;
typedef __attribute__((ext_vector_type(8))) float v8f;

#define NROWS 200000
#define NIN   64
#define NS    100
#define NS_PAD 112          // 7 tiles of 16 sample-columns, zero-padded
#define WAVES_PER_BLOCK 5   // 5 waves * 2500 blocks = 12500 waves = NROWS/16 exactly

__global__ __launch_bounds__(32 * WAVES_PER_BLOCK)
void mf_logreg_wmma_kernel(const float* __restrict__ X,
                           const float* __restrict__ w_mu,
                           const float* __restrict__ w_log_var,
                           const float* __restrict__ z,
                           float* __restrict__ out)
{
    __shared__ float s_wmu [NIN];                 // w_mu staged
    __shared__ float s_expw[NIN];                 // exp(w_log_var) staged
    __shared__ float s_z   [NS_PAD];              // z, zero-padded to 112
    __shared__ float s_mean[WAVES_PER_BLOCK][16];
    __shared__ float s_std [WAVES_PER_BLOCK][16];

    const int tid  = threadIdx.x;
    const int lane = tid & 31;
    const int wid  = tid >> 5;
    const int tile = blockIdx.x * WAVES_PER_BLOCK + wid;   // always < 12500
    const int r0   = tile * 16;

    // ---- Stage tiny shared operands once per block (all 160 threads reach) ----
    if (tid < NIN) {
        s_wmu [tid] = w_mu[tid];
        s_expw[tid] = __expf(w_log_var[tid]);
    }
    if (tid < NS_PAD) {
        s_z[tid] = (tid < NS) ? z[tid] : 0.0f;
    }
    __syncthreads();

    const int   mrow    = lane & 15;       // A row / B,C,D column this lane covers
    const int   khalf   = (lane >> 4) * 2; // lanes 16-31 hold K=2,3 of each 4-chunk
    const float colMask = (mrow == 0) ? 1.0f : 0.0f;  // only B column 0 non-zero

    // ---------------- Phase 1: mean_i, var_i via WMMA K-chunks ----------------
    v8f acc_mu  = {};
    v8f acc_var = {};

    const float* xrow = X + (r0 + mrow) * NIN;

    #pragma unroll
    for (int c = 0; c < 16; ++c) {
        const int kb = 4 * c + khalf;

        const float2 av = *reinterpret_cast<const float2*>(xrow + kb);
        v2f a;  a.x  = av.x;        a.y  = av.y;
        v2f a2; a2.x = av.x * av.x; a2.y = av.y * av.y;

        // Branch-free B fragments: broadcast LDS reads, masked by column-0 lane.
        const float2 wv = *reinterpret_cast<const float2*>(&s_wmu [kb]);
        const float2 ev = *reinterpret_cast<const float2*>(&s_expw[kb]);
        v2f bmu; bmu.x = wv.x * colMask; bmu.y = wv.y * colMask;
        v2f bvr; bvr.x = ev.x * colMask; bvr.y = ev.y * colMask;

        acc_mu  = __builtin_amdgcn_wmma_f32_16x16x4_f32(false, a,  false, bmu,
                                                        (short)0, acc_mu,  false, false);
        acc_var = __builtin_amdgcn_wmma_f32_16x16x4_f32(false, a2, false, bvr,
                                                        (short)0, acc_var, false, false);
    }

    // Column 0 of D lives in lane 0 (M=0..7 across VGPRs 0..7) and lane 16 (M=8..15).
    if (mrow == 0) {
        const int mbase = (lane >> 4) * 8;
        #pragma unroll
        for (int v = 0; v < 8; ++v) {
            s_mean[wid][mbase + v] = acc_mu[v];
            s_std [wid][mbase + v] = sqrtf(acc_var[v]);
        }
    }
    __syncthreads();   // all 5 waves always reach this (exact tiling, no early exit)

    // ---------------- Phase 2: activation tiles via rank-2 WMMA ----------------
    // A: lanes 0-15 carry (std_M, mean_M) in K=0,1 ; lanes 16-31 carry K=2,3 = 0.
    const float loMask = (lane < 16) ? 1.0f : 0.0f;
    v2f aSM;
    aSM.x = s_std [wid][mrow] * loMask;
    aSM.y = s_mean[wid][mrow] * loMask;

    float* __restrict__ Y   = out;
    float* __restrict__ ACT = out + NROWS * NS;

    const v8f zeroC = {};
    const int rbase = r0 + ((lane >> 4) * 8);

    #pragma unroll
    for (int t = 0; t < 7; ++t) {
        const int s0   = t * 16;
        const int sidx = s0 + mrow;          // this lane's sample column

        // B: row 0 = z[s0..s0+15] (zero-padded), row 1 = ones, rows 2,3 = 0.
        v2f b;
        b.x = s_z[sidx] * loMask;
        b.y = loMask;

        v8f d = __builtin_amdgcn_wmma_f32_16x16x4_f32(false, aSM, false, b,
                                                      (short)0, zeroC, false, false);

        // D[M][N] = std_M * z[s0+N] + mean_M.  lane -> column N=mrow,
        // VGPR v -> row M = v + 8*(lane>=16).  Stores: 64B-contiguous per half-wave.
        if (sidx < NS) {
            #pragma unroll
            for (int v = 0; v < 8; ++v) {
                const int off  = (rbase + v) * NS + sidx;
                const float act = d[v];
                ACT[off] = act;
                Y[off]   = __builtin_amdgcn_rcpf(1.0f + __expf(-act));
            }
        }
    }
}

extern "C" void kernel_launch(void* const* d_in, const int* in_sizes, int n_in,
                              void* d_out, int out_size, void* d_ws, size_t ws_size,
                              hipStream_t stream) {
    (void)in_sizes; (void)n_in; (void)out_size; (void)d_ws; (void)ws_size;
    const float* X         = (const float*)d_in[0];
    const float* w_mu      = (const float*)d_in[1];
    const float* w_log_var = (const float*)d_in[2];
    const float* z         = (const float*)d_in[3];
    float* out             = (float*)d_out;

    const int tiles  = NROWS / 16;                    // 12500
    const int blocks = tiles / WAVES_PER_BLOCK;       // 2500 (exact)
    mf_logreg_wmma_kernel<<<blocks, 32 * WAVES_PER_BLOCK, 0, stream>>>(
        X, w_mu, w_log_var, z, out);
}